// SumAggregator_19292993094232
// MI455X (gfx1250) — compile-verified
//
#include <hip/hip_runtime.h>

// Problem constants (from reference)
#define T_ROWS  16384
#define D_COLS  512
#define N_SPANS 262144
#define CHUNK   128
#define NCHUNK  (T_ROWS / CHUNK)   // 128

typedef __attribute__((ext_vector_type(2))) float v2f;
typedef __attribute__((ext_vector_type(4))) float v4f;
typedef __attribute__((ext_vector_type(8))) float v8f;

// ---------------------------------------------------------------------------
// Pass 1: per-chunk column sums.  grid = NCHUNK blocks, 512 threads.
// Thread j sums column j over the chunk's 128 rows (fully coalesced loads).
// ---------------------------------------------------------------------------
__global__ __launch_bounds__(512)
void chunk_sum_kernel(const float* __restrict__ emb, float* __restrict__ chunkSum) {
    const int c = blockIdx.x;
    const int j = threadIdx.x;
    const float* p = emb + (size_t)c * CHUNK * D_COLS + j;
    float s = 0.f;
    #pragma unroll 8
    for (int r = 0; r < CHUNK; ++r) s += p[(size_t)r * D_COLS];
    chunkSum[c * D_COLS + j] = s;
}

// ---------------------------------------------------------------------------
// Pass 2: exclusive scan of the 128 chunk sums per column; also writes
// csum row 0 = 0.  1 block, 512 threads (thread j owns column j).
// ---------------------------------------------------------------------------
__global__ __launch_bounds__(512)
void chunk_scan_kernel(const float* __restrict__ chunkSum,
                       float* __restrict__ chunkPrefix,
                       float* __restrict__ csum) {
    const int j = threadIdx.x;
    csum[j] = 0.f;                       // csum[0][:] = 0
    float run = 0.f;
    for (int c = 0; c < NCHUNK; ++c) {
        chunkPrefix[c * D_COLS + j] = run;
        run += chunkSum[c * D_COLS + j];
    }
}

// ---------------------------------------------------------------------------
// Pass 3: in-chunk inclusive scan via WMMA.
//   csum[chunk*128 + i + 1][col] = chunkPrefix[chunk][col] + sum_{r<=i} X[r][col]
// 16-row tile scan = L(16x16 lower-tri ones) * X(16x16), done as a chain of
// four V_WMMA_F32_16X16X4_F32 with the running carry held in the C operand.
// grid = NCHUNK blocks, 256 threads (8 waves); each wave owns 4 col-tiles.
// EXEC is all-ones everywhere WMMA executes (uniform control flow).
// ---------------------------------------------------------------------------
__global__ __launch_bounds__(256)
void csum_wmma_kernel(const float* __restrict__ emb,
                      const float* __restrict__ chunkPrefix,
                      float* __restrict__ csum) {
    const int chunk = blockIdx.x;
    const int lane  = threadIdx.x & 31;
    const int wave  = threadIdx.x >> 5;       // 0..7
    const int n     = lane & 15;              // col within tile; also A-matrix row M
    const int half  = lane >> 4;              // K half-select per ISA f32 layouts

    for (int ct = wave; ct < D_COLS / 16; ct += 8) {
        const int col = ct * 16 + n;
        float carry = chunkPrefix[chunk * D_COLS + col];   // inter-chunk prefix

        for (int rt = 0; rt < CHUNK / 16; ++rt) {
            const int rowbase = chunk * CHUNK + rt * 16;

            // C = carry broadcast down each column (adds carry to every row)
            v8f c;
            #pragma unroll
            for (int v = 0; v < 8; ++v) c[v] = carry;

            #pragma unroll
            for (int k = 0; k < 4; ++k) {
                const int j0 = 4 * k + 2 * half;   // this lane's K pair
                // A = columns j0, j0+1 of lower-triangular ones: L[m][j] = (j<=m)
                v2f a;
                a.x = (j0     <= n) ? 1.f : 0.f;
                a.y = (j0 + 1 <= n) ? 1.f : 0.f;
                // B = rows j0, j0+1 of the 16x16 data tile, column `col`
                const float* src = emb + (size_t)(rowbase + j0) * D_COLS + col;
                v2f b;
                b.x = src[0];
                b.y = src[D_COLS];
                // D = A*B + C   (chained accumulate over K)
                c = __builtin_amdgcn_wmma_f32_16x16x4_f32(
                        /*neg_a=*/false, a, /*neg_b=*/false, b,
                        /*c_mod=*/(short)0, c, /*reuse_a=*/false, /*reuse_b=*/false);
            }

            // C/D layout: lanes 0-15 hold rows 0..7, lanes 16-31 rows 8..15
            float* dst = csum + (size_t)(rowbase + half * 8 + 1) * D_COLS + col;
            #pragma unroll
            for (int v = 0; v < 8; ++v) dst[(size_t)v * D_COLS] = c[v];

            // new carry = last scan row (row 15, col n) = lane 16+n, VGPR 7
            carry = __shfl(c[7], 16 + n, 32);
        }
    }
}

// ---------------------------------------------------------------------------
// Pass 4: span gather.  out[s][:] = csum[end+1][:] - csum[start][:]
// 2 spans per 256-thread block; 128 lanes * float4 = one 512-float row.
// csum (33.5 MB) stays L2-resident; output stores are non-temporal so the
// 512 MB stream does not evict it from the 192 MB L2.
// ---------------------------------------------------------------------------
__global__ __launch_bounds__(256)
void span_gather_kernel(const int* __restrict__ spans,
                        const float* __restrict__ csum,
                        float* __restrict__ out) {
    const int s = blockIdx.x * 2 + (threadIdx.x >> 7);
    const int t = threadIdx.x & 127;
    const int start = spans[2 * s + 0];
    const int end   = spans[2 * s + 1];
    const v4f* pe = (const v4f*)(csum + (size_t)(end + 1) * D_COLS);
    const v4f* ps = (const v4f*)(csum + (size_t)start * D_COLS);
    v4f r = pe[t] - ps[t];
    __builtin_nontemporal_store(r, (v4f*)out + (size_t)s * (D_COLS / 4) + t);
}

// ---------------------------------------------------------------------------
extern "C" void kernel_launch(void* const* d_in, const int* in_sizes, int n_in,
                              void* d_out, int out_size, void* d_ws, size_t ws_size,
                              hipStream_t stream) {
    const float* emb   = (const float*)d_in[0];     // 16384 x 512 f32
    const int*   spans = (const int*)d_in[1];       // 262144 x 2 i32
    float*       out   = (float*)d_out;             // 262144 x 512 f32

    // Workspace layout (all f32):
    //   csum        : (T_ROWS+1) * D_COLS          = 33,556,480 B
    //   chunkSum    : NCHUNK * D_COLS              =    262,144 B
    //   chunkPrefix : NCHUNK * D_COLS              =    262,144 B
    float* csum        = (float*)d_ws;
    float* chunkSum    = csum + (size_t)(T_ROWS + 1) * D_COLS;
    float* chunkPrefix = chunkSum + (size_t)NCHUNK * D_COLS;

    chunk_sum_kernel<<<NCHUNK, 512, 0, stream>>>(emb, chunkSum);
    chunk_scan_kernel<<<1, 512, 0, stream>>>(chunkSum, chunkPrefix, csum);
    csum_wmma_kernel<<<NCHUNK, 256, 0, stream>>>(emb, chunkPrefix, csum);
    span_gather_kernel<<<N_SPANS / 2, 256, 0, stream>>>(spans, csum, out);
}